// NeuralNetwork_37838661878387
// MI455X (gfx1250) — compile-verified
//
#include <hip/hip_runtime.h>
#include <math.h>

#define NN    8192
#define NL    16
#define NE    (8192 * 128)      // edges per hidden layer
#define NOUT  16
#define EOUT  (8192 * 16)       // edges into output layer
#define OUT_BLOCKS 64

typedef __attribute__((ext_vector_type(2))) float v2f;
typedef __attribute__((ext_vector_type(8))) float v8f;

__device__ __forceinline__ float lrelu(float v) { return v >= 0.0f ? v : 0.01f * v; }

// One hidden layer: acc[dst] += act(vin[src]) * w   (acc pre-zeroed)
// 1024 blocks x 256 threads, 4 edges/thread via B128 streaming loads.
__global__ __launch_bounds__(256) void layer_kernel(
    const float* __restrict__ vin, const int* __restrict__ src,
    const int* __restrict__ dst, const float* __restrict__ w,
    float* __restrict__ acc, int apply_act)
{
    int gid = blockIdx.x * 256 + threadIdx.x;
    int4   s4 = ((const int4*)src)[gid];
    int4   d4 = ((const int4*)dst)[gid];
    float4 w4 = ((const float4*)w)[gid];
    float v0 = vin[s4.x], v1 = vin[s4.y], v2 = vin[s4.z], v3 = vin[s4.w];
    if (apply_act) { v0 = lrelu(v0); v1 = lrelu(v1); v2 = lrelu(v2); v3 = lrelu(v3); }
    atomicAdd(&acc[d4.x], v0 * w4.x);
    atomicAdd(&acc[d4.y], v1 * w4.y);
    atomicAdd(&acc[d4.z], v2 * w4.z);
    atomicAdd(&acc[d4.w], v3 * w4.w);
}

// Output edges -> per-block 16 partial sums (per-wave LDS bins, no global atomics).
// 64 blocks x 256 threads x 8 edges = 131072 edges.
__global__ __launch_bounds__(256) void out_kernel(
    const float* __restrict__ vin, const int* __restrict__ src,
    const int* __restrict__ dst, const float* __restrict__ w,
    float* __restrict__ partials)
{
    __shared__ float bins[8][NOUT];
    int tid  = threadIdx.x;
    int wave = tid >> 5;
    if (tid < 8 * NOUT) ((float*)bins)[tid] = 0.0f;
    __syncthreads();

    int base = (blockIdx.x * 256 + tid) * 2;   // vec4 index; 2 x 4 edges per thread
    #pragma unroll
    for (int i = 0; i < 2; ++i) {
        int4   s4 = ((const int4*)src)[base + i];
        int4   d4 = ((const int4*)dst)[base + i];
        float4 w4 = ((const float4*)w)[base + i];
        atomicAdd(&bins[wave][d4.x], lrelu(vin[s4.x]) * w4.x);
        atomicAdd(&bins[wave][d4.y], lrelu(vin[s4.y]) * w4.y);
        atomicAdd(&bins[wave][d4.z], lrelu(vin[s4.z]) * w4.z);
        atomicAdd(&bins[wave][d4.w], lrelu(vin[s4.w]) * w4.w);
    }
    __syncthreads();
    if (tid < NOUT) {
        float s = 0.0f;
        #pragma unroll
        for (int wv = 0; wv < 8; ++wv) s += bins[wv][tid];
        partials[blockIdx.x * NOUT + tid] = s;
    }
}

// Reduce 64 groups of 16 partials with V_WMMA_F32_16X16X4_F32 (B = ones),
// then leaky-ReLU + softmax over 16. One wave; EXEC all-ones through the WMMA loop.
__global__ __launch_bounds__(32) void final_kernel(
    const float* __restrict__ partials, float* __restrict__ out)
{
    __shared__ float tmp[NOUT];
    int lane = threadIdx.x;
    int m  = lane & 15;             // A-matrix row (output index)
    int kb = (lane >> 4) * 2;       // lanes 16-31 hold K=2,3

    v8f c = {};
    v2f b; b.x = 1.0f; b.y = 1.0f;  // all-ones B: D[m][n] = sum_k A[m][k] + C

    for (int g = 0; g < OUT_BLOCKS; g += 4) {
        v2f a;
        a.x = partials[(g + kb + 0) * NOUT + m];
        a.y = partials[(g + kb + 1) * NOUT + m];
        c = __builtin_amdgcn_wmma_f32_16x16x4_f32(
                false, a, false, b, (short)0, c, false, false);
    }

    // D layout: lane 0 VGPR r -> D[r][0] (outputs 0..7); lane 16 VGPR r -> D[8+r][0]
    if (lane == 0)  { for (int r = 0; r < 8; ++r) tmp[r]     = c[r]; }
    if (lane == 16) { for (int r = 0; r < 8; ++r) tmp[8 + r] = c[r]; }
    __syncthreads();

    if (lane == 0) {
        float o[NOUT];
        float mx = -INFINITY;
        for (int i = 0; i < NOUT; ++i) { o[i] = lrelu(tmp[i]); mx = fmaxf(mx, o[i]); }
        float sum = 0.0f;
        for (int i = 0; i < NOUT; ++i) { o[i] = expf(o[i] - mx); sum += o[i]; }
        float inv = 1.0f / sum;
        for (int i = 0; i < NOUT; ++i) out[i] = o[i] * inv;
    }
}

extern "C" void kernel_launch(void* const* d_in, const int* in_sizes, int n_in,
                              void* d_out, int out_size, void* d_ws, size_t ws_size,
                              hipStream_t stream)
{
    (void)in_sizes; (void)n_in; (void)out_size; (void)ws_size;
    const float* x        = (const float*)d_in[0];
    const float* edge_w   = (const float*)d_in[1];
    const float* out_w    = (const float*)d_in[2];
    const int*   edge_src = (const int*)d_in[3];
    const int*   edge_dst = (const int*)d_in[4];
    const int*   out_src  = (const int*)d_in[5];
    const int*   out_dst  = (const int*)d_in[6];
    float* out = (float*)d_out;

    float* buf0     = (float*)d_ws;          // raw (pre-activation) values, double-buffered
    float* buf1     = buf0 + NN;
    float* partials = buf1 + NN;             // OUT_BLOCKS * 16 floats

    for (int l = 0; l < NL; ++l) {
        float* acc       = (l & 1) ? buf1 : buf0;
        const float* vin = (l == 0) ? x : ((l & 1) ? buf0 : buf1);
        hipMemsetAsync(acc, 0, NN * sizeof(float), stream);
        layer_kernel<<<NE / 1024, 256, 0, stream>>>(
            vin, edge_src + (size_t)l * NE, edge_dst + (size_t)l * NE,
            edge_w + (size_t)l * NE, acc, l != 0);
    }
    // after l=15 (odd), final raw hidden values live in buf1
    out_kernel<<<OUT_BLOCKS, 256, 0, stream>>>(buf1, out_src, out_dst, out_w, partials);
    final_kernel<<<1, 32, 0, stream>>>(partials, out);
}